// SubsurfaceFlowTerm_78065325572396
// MI455X (gfx1250) — compile-verified
//
#include <hip/hip_runtime.h>

typedef __attribute__((ext_vector_type(2))) float v2f;
typedef __attribute__((ext_vector_type(8))) float v8f;

// SWAPX16 ds_swizzle: lane L receives value from lane L^16.
__device__ __forceinline__ float swapx16(float v) {
    return __int_as_float(__builtin_amdgcn_ds_swizzle(__float_as_int(v), 0x401f));
}

__device__ __forceinline__ float fast_tanh(float x) {
#if __has_builtin(__builtin_amdgcn_tanhf)
    return __builtin_amdgcn_tanhf(x);          // v_tanh_f32 on gfx1250
#else
    float e = __expf(2.0f * x);
    return 1.0f - 2.0f / (e + 1.0f);
#endif
}

__device__ __forceinline__ float fast_exp2(float x) {
#if __has_builtin(__builtin_amdgcn_exp2f)
    return __builtin_amdgcn_exp2f(x);          // v_exp_f32
#else
    return exp2f(x);
#endif
}

__device__ __forceinline__ float fast_log2(float x) {
#if __has_builtin(__builtin_amdgcn_logf)
    return __builtin_amdgcn_logf(x);           // v_log_f32
#else
    return log2f(x);
#endif
}

// Branchless guarded load: unconditional load from a clamped index + select.
__device__ __forceinline__ float ldsel(bool p, const float* ptr, int idx) {
    float v = ptr[p ? idx : 0];
    return p ? v : 0.0f;
}

__global__ __launch_bounds__(256) void hydro_mlp_wmma(
    const float* __restrict__ x,  const float* __restrict__ S1,
    const float* __restrict__ W1, const float* __restrict__ b1,
    const float* __restrict__ W2, const float* __restrict__ b2,
    const float* __restrict__ W3, const float* __restrict__ b3,
    float* __restrict__ out, int n)
{
    const int  lane = threadIdx.x & 31;
    const int  ln   = lane & 15;         // batch column within half
    const int  lh   = lane >> 4;         // lane half
    const bool lo   = (lh == 0);
    const bool m0v  = (ln < 6);          // valid MLP0 hidden row (M = 0..5)
    const bool m1v  = (ln >= 8 && ln < 14); // valid MLP1 hidden row (M = 8..13)
    const int  jm1  = ln - 8;

    const int wavesPerBlk = blockDim.x >> 5;
    const int wave   = blockIdx.x * wavesPerBlk + (threadIdx.x >> 5);
    const int nWaves = gridDim.x * wavesPerBlk;
    const int numTiles = (n + 15) >> 4;

    const int d0 = 2 * lh, d1 = 2 * lh + 1;   // K indices held by this lane half

    // ================= Weight preload (all branchless) =================
    // --- Combined MLP0 (M rows 0-5) + MLP1 (M rows 8-13) ---
    // Layer 1 A (16x4): A[j, d] = W1[k][d][j_local]
    v2f A1c;
    {
        float w0x = ldsel(m0v, W1,      d0 * 6 + ln);
        float w0y = ldsel(m0v, W1,      d1 * 6 + ln);
        float w1x = ldsel(m1v, W1, 24 + d0 * 6 + jm1);
        float w1y = ldsel(m1v, W1, 24 + d1 * 6 + jm1);
        A1c[0] = m0v ? w0x : w1x;
        A1c[1] = m0v ? w0y : w1y;
    }
    // Layer 2 block-diagonal, K = 12 split into 3 chunks of 4.
    v2f A2c0, A2c1, A2c2;
    {
        // chunk0: K=0..3 -> MLP0 i=0..3 (rows 0-5 only)
        A2c0[0] = ldsel(m0v, W2, d0 * 6 + ln);
        A2c0[1] = ldsel(m0v, W2, d1 * 6 + ln);
        // chunk1: K=4..7 -> lh0: MLP0 i=4,5 (rows 0-5); lh1: MLP1 i=0,1 (rows 8-13)
        float c1x0 = ldsel(m0v, W2, 4 * 6 + ln);
        float c1y0 = ldsel(m0v, W2, 5 * 6 + ln);
        float c1x1 = ldsel(m1v, W2, 36 + 0 * 6 + jm1);
        float c1y1 = ldsel(m1v, W2, 36 + 1 * 6 + jm1);
        A2c1[0] = lo ? c1x0 : c1x1;
        A2c1[1] = lo ? c1y0 : c1y1;
        // chunk2: K=8..11 -> MLP1 i=2..5 (rows 8-13 only)
        A2c2[0] = ldsel(m1v, W2, 36 + (d0 + 2) * 6 + jm1);
        A2c2[1] = ldsel(m1v, W2, 36 + (d1 + 2) * 6 + jm1);
    }
    // Per-lane-half constants for the combined chain (MLP0 in lanes 0-15, MLP1 in 16-31)
    float bb1[6], bb2[6], ww3[6];
#pragma unroll
    for (int j = 0; j < 6; ++j) {
        bb1[j] = lo ? b1[j] : b1[6 + j];
        bb2[j] = lo ? b2[j] : b2[6 + j];
        ww3[j] = lo ? W3[j] : W3[6 + j];
    }
    const float bb3  = lo ? b3[0]  : b3[1];
    const float lowc = lo ? 100.0f : 0.01f;   // S1max | ks bounds
    const float rngc = lo ? 400.0f : 99.99f;

    // --- MLP2 ("n" parameter), rows 0-5 only ---
    v2f a1_2, a2a_2, a2b_2;
    {
        a1_2[0]  = ldsel(m0v, W1, 48 + d0 * 6 + ln);
        a1_2[1]  = ldsel(m0v, W1, 48 + d1 * 6 + ln);
        a2a_2[0] = ldsel(m0v, W2, 72 + d0 * 6 + ln);
        a2a_2[1] = ldsel(m0v, W2, 72 + d1 * 6 + ln);
        a2b_2[0] = ldsel(m0v && lo, W2, 72 + 4 * 6 + ln);
        a2b_2[1] = ldsel(m0v && lo, W2, 72 + 5 * 6 + ln);
    }
    float b1_2[6], b2_2[6], w3_2[6];
#pragma unroll
    for (int j = 0; j < 6; ++j) {        // uniform -> SGPRs
        b1_2[j] = b1[12 + j];
        b2_2[j] = b2[12 + j];
        w3_2[j] = W3[12 + j];
    }
    const float b3_2 = b3[2];

    // ========================= Tile loop =========================
    for (int t = wave; t < numTiles; t += nWaves) {
        const int m0 = t << 4;
        int row = m0 + ln; if (row > n - 1) row = n - 1;

        // B = X^T (4x16): lane holds N=ln, K = d0,d1  -> one b64 load/lane
        v2f bx = *(const v2f*)(x + (size_t)row * 4 + d0);

        if (t + nWaves < numTiles) {
            __builtin_prefetch(x + ((size_t)m0 + ((size_t)nWaves << 4)) * 4 + lane * 8, 0, 1);
        }

        v8f z = {};

        // ---------- Combined MLP0+MLP1 ----------
        v8f h1c = __builtin_amdgcn_wmma_f32_16x16x4_f32(
            false, A1c, false, bx, (short)0, z, false, false);
#pragma unroll
        for (int j = 0; j < 6; ++j)
            h1c[j] = fast_tanh(h1c[j] + bb1[j]);

        // Swizzles hoisted & unconditional -> pure v_cndmask selects below.
        float sw2 = swapx16(h1c[2]);
        float sw3 = swapx16(h1c[3]);
        v2f B0, B1, B2;
        B0[0] = lo ? h1c[0] : sw2;   // K=0 | K=2
        B0[1] = lo ? h1c[1] : sw3;   // K=1 | K=3
        B1[0] = lo ? h1c[4] : h1c[0];// K=4 | K=6 (MLP1 i=0, already in hi lanes)
        B1[1] = lo ? h1c[5] : h1c[1];// K=5 | K=7
        B2[0] = lo ? sw2 : h1c[4];   // K=8 (MLP1 i=2) | K=10 (MLP1 i=4)
        B2[1] = lo ? sw3 : h1c[5];   // K=9 | K=11

        v8f h2c = __builtin_amdgcn_wmma_f32_16x16x4_f32(
            false, A2c0, false, B0, (short)0, z, false, false);
        h2c = __builtin_amdgcn_wmma_f32_16x16x4_f32(
            false, A2c1, false, B1, (short)0, h2c, false, false);
        h2c = __builtin_amdgcn_wmma_f32_16x16x4_f32(
            false, A2c2, false, B2, (short)0, h2c, false, false);

        float y01 = bb3;
#pragma unroll
        for (int j = 0; j < 6; ++j)
            y01 += ww3[j] * fast_tanh(h2c[j] + bb2[j]);
        // lanes 0-15: S1max ; lanes 16-31: ks
        float val01 = lowc + rngc / (1.0f + __expf(-y01));

        // ---------- MLP2 ("n") ----------
        v8f h1 = __builtin_amdgcn_wmma_f32_16x16x4_f32(
            false, a1_2, false, bx, (short)0, z, false, false);
#pragma unroll
        for (int j = 0; j < 6; ++j)
            h1[j] = fast_tanh(h1[j] + b1_2[j]);

        float s2 = swapx16(h1[2]);
        float s3 = swapx16(h1[3]);
        v2f B2a, B2b;
        B2a[0] = lo ? h1[0] : s2;
        B2a[1] = lo ? h1[1] : s3;
        B2b[0] = lo ? h1[4] : 0.0f;
        B2b[1] = lo ? h1[5] : 0.0f;

        v8f h2 = __builtin_amdgcn_wmma_f32_16x16x4_f32(
            false, a2a_2, false, B2a, (short)0, z, false, false);
        h2 = __builtin_amdgcn_wmma_f32_16x16x4_f32(
            false, a2b_2, false, B2b, (short)0, h2, false, false);

        float y2 = b3_2;
#pragma unroll
        for (int j = 0; j < 6; ++j)
            y2 += w3_2[j] * fast_tanh(h2[j] + b2_2[j]);
        float val2 = 0.01f + 9.99f / (1.0f + __expf(-y2));   // n

        // ---------- epilogue: flow = clip(ks * (S1/S1max)^n, 0, S1max) ----------
        float ksv  = swapx16(val01);     // bring ks (hi lanes) down to lanes 0-15
        float s1   = S1[row];
        float r    = s1 / val01;         // lanes 0-15: S1 / S1max
        float flow = ksv * fast_exp2(val2 * fast_log2(r));
        flow = fminf(fmaxf(flow, 0.0f), val01);

        if (lo && (m0 + ln) < n)
            out[m0 + ln] = flow;
    }
}

extern "C" void kernel_launch(void* const* d_in, const int* in_sizes, int n_in,
                              void* d_out, int out_size, void* d_ws, size_t ws_size,
                              hipStream_t stream) {
    const float* x  = (const float*)d_in[0];
    const float* S1 = (const float*)d_in[1];
    const float* W1 = (const float*)d_in[2];
    const float* b1 = (const float*)d_in[3];
    const float* W2 = (const float*)d_in[4];
    const float* b2 = (const float*)d_in[5];
    const float* W3 = (const float*)d_in[6];
    const float* b3 = (const float*)d_in[7];
    float* out = (float*)d_out;

    const int n = in_sizes[1];           // S1 flat count == N rows
    const int numTiles = (n + 15) / 16;
    const int wavesPerBlk = 256 / 32;
    int grid = (numTiles + wavesPerBlk - 1) / wavesPerBlk;
    if (grid > 2048) grid = 2048;        // grid-stride: amortize weight preload
    if (grid < 1) grid = 1;

    hipLaunchKernelGGL(hydro_mlp_wmma, dim3(grid), dim3(256), 0, stream,
                       x, S1, W1, b1, W2, b2, W3, b3, out, n);
}